// KAN_38293928411788
// MI455X (gfx1250) — compile-verified
//
#include <hip/hip_runtime.h>

// ---------------------------------------------------------------------------
// KAN (2-layer spline network) for gfx1250.
//   out[n,o] = sum_{i,j} prod_d clip((x[n,i]-k[o,i,j])*invd[o,i,j,d],0,1) * c[o,i,j]
// Reduction over k=(o',j) (96 values per input feature) runs on the matrix
// pipe via V_WMMA_F32_16X16X4_F32. Coefficients are folded into the A operand
// (record side), so the B operand is a loop-invariant 0/1 delta pattern held
// in registers. LDS staging uses GLOBAL_LOAD_ASYNC_TO_LDS.
// ---------------------------------------------------------------------------

typedef float v2f __attribute__((ext_vector_type(2)));
typedef float v8f __attribute__((ext_vector_type(8)));
typedef int   v4i __attribute__((vector_size(16)));

#define NK 10   // knots per edge
#define KJ 6    // J = NK - 1 - DEGREE

#if __has_builtin(__builtin_amdgcn_global_load_async_to_lds_b128)
#define HAVE_ASYNC_LDS 1
#else
#define HAVE_ASYNC_LDS 0
#endif

#define AS_G __attribute__((address_space(1)))
#define AS_L __attribute__((address_space(3)))

#if HAVE_ASYNC_LDS
__device__ __forceinline__ void async_b128(void* g, void* l) {
  __builtin_amdgcn_global_load_async_to_lds_b128((AS_G v4i*)g, (AS_L v4i*)l,
                                                 0, 0);
}
#endif

__device__ __forceinline__ float clamp01(float v) {
  return __builtin_fminf(__builtin_fmaxf(v, 0.0f), 1.0f);
}

// Two A-operand values (K-pair) for one row, coeff folded in.
// q0 = {kj0,kj1,inv0_0,inv0_1}, q1 = {inv1_0,inv1_1,inv2_0,inv2_1},
// q2 = {cf0,cf1,pad,pad}.
__device__ __forceinline__ v2f basis_cf_pair(float xv, float4 q0, float4 q1,
                                             float4 q2) {
  float n0 = xv - q0.x, n1 = xv - q0.y;
  float t0 = clamp01(n0 * q0.z) * clamp01(n0 * q1.x) * clamp01(n0 * q1.z);
  float t1 = clamp01(n1 * q0.w) * clamp01(n1 * q1.y) * clamp01(n1 * q1.w);
  v2f r; r.x = t0 * q2.x; r.y = t1 * q2.y;
  return r;
}

// Build pair-interleaved per-(o,i) records (12 floats per K-pair).
// kk = oi*6 + j within a 16-wide o-block; pair p = kk>>1, parity = kk&1:
//   floats[12] = {kj0,kj1, i0_0,i0_1, i1_0,i1_1, i2_0,i2_1, cf0,cf1, pad,pad}
__global__ void kan_prep(const float* __restrict__ knots,
                         const float* __restrict__ coeffs,
                         float*  __restrict__ recA,   // (O/16, I, 48 pairs, 12)
                         int O, int I) {
  int idx = blockIdx.x * blockDim.x + threadIdx.x;
  int tot = O * I * KJ;
  if (idx >= tot) return;
  int j    = idx % KJ;
  int rest = idx / KJ;
  int i    = rest % I;
  int o    = rest / I;
  const float* k = knots + ((size_t)o * I + i) * NK;
  float kj = k[j];
  float i0 = 1.0f / (k[j + 1] - kj);
  float i1 = 1.0f / (k[j + 2] - kj);
  float i2 = 1.0f / (k[j + 3] - kj);
  float cf = coeffs[((size_t)o * I + i) * NK + j];
  int ob = o >> 4, oi = o & 15;
  int kk = oi * KJ + j;
  int p = kk >> 1, par = kk & 1;
  float* rb = recA + ((size_t)(ob * I + i) * 48 + p) * 12;
  rb[0 + par] = kj;
  rb[2 + par] = i0;
  rb[4 + par] = i1;
  rb[6 + par] = i2;
  rb[8 + par] = cf;
}

// One wave per workgroup computes a 32n x 16o output tile (two 16x16 WMMA
// accumulators sharing staged records and the delta B pattern).
// K-space per input feature: k = o'*6 + j in [0,96) -> 24 K-steps of 4.
// waves_per_eu(4) caps the VGPR budget at ~256 so no s_set_vgpr_msb traffic.
template <int O, int I, int IC>
__global__ __launch_bounds__(32)
__attribute__((amdgpu_waves_per_eu(4))) void kan_layer(
    const float*  __restrict__ x,     // N x I
    const float4* __restrict__ recA,  // pair-interleaved records (3 float4/pair)
    float*        __restrict__ out) { // N x O
  __shared__ float4 sR[IC * 144];     // 48 pairs * 3 float4 per feature

  const int lane = threadIdx.x;       // full wave, EXEC all ones
  const int m    = lane & 15;         // row M (A) / col N (B,C)
  const int lh   = lane >> 4;         // K-pair selector {2lh, 2lh+1}
  const int nb   = blockIdx.x * 32;   // n-tile base (two 16-row tiles)
  const int ob   = blockIdx.y;        // o-block (16 outputs)

  const float4* gR = recA + (size_t)ob * I * 144;

  // Loop-invariant delta B pattern: b[s] = {delta(o'(kk0)==m), delta(o'(kk1)==m)}
  // with kk0 = 4s + 2*lh, kk1 = kk0 + 1.
  v2f bp[24];
#pragma unroll
  for (int s = 0; s < 24; ++s) {
    const int oo0 = lh ? (4 * s + 2) / 6 : (4 * s) / 6;
    const int oo1 = lh ? (4 * s + 3) / 6 : (4 * s + 1) / 6;
    bp[s].x = (oo0 == m) ? 1.0f : 0.0f;
    bp[s].y = (oo1 == m) ? 1.0f : 0.0f;
  }

  v8f acc0 = {0.f, 0.f, 0.f, 0.f, 0.f, 0.f, 0.f, 0.f};
  v8f acc1 = acc0;

#pragma unroll 1
  for (int i0 = 0; i0 < I; i0 += IC) {
    // ---- stage records for IC input features into LDS ----
#if HAVE_ASYNC_LDS
    asm volatile("s_wait_dscnt 0x0" ::: "memory");  // prior chunk consumed
    for (int t = lane; t < IC * 144; t += 32) {
      async_b128((void*)(gR + (size_t)i0 * 144 + t), (void*)(sR + t));
    }
    asm volatile("s_wait_asynccnt 0x0" ::: "memory");
#else
    for (int t = lane; t < IC * 144; t += 32) {  // single wave: DS in-order
      sR[t] = gR[(size_t)i0 * 144 + t];
    }
#endif

    // ---- hoisted x loads: IC features for both n-tiles ----
    float xr0[IC], xr1[IC];
    {
      const float4* xp0 = (const float4*)&x[(size_t)(nb + m) * I + i0];
      const float4* xp1 = (const float4*)&x[(size_t)(nb + 16 + m) * I + i0];
#pragma unroll
      for (int q = 0; q < IC / 4; ++q) {
        float4 a = xp0[q], b = xp1[q];
        xr0[4 * q + 0] = a.x; xr0[4 * q + 1] = a.y;
        xr0[4 * q + 2] = a.z; xr0[4 * q + 3] = a.w;
        xr1[4 * q + 0] = b.x; xr1[4 * q + 1] = b.y;
        xr1[4 * q + 2] = b.z; xr1[4 * q + 3] = b.w;
      }
    }

#pragma unroll 1
    for (int ii = 0; ii < IC; ++ii) {
      // lane's record base: pair index (2s + lh)*3 float4s from feature base
      const float4* rr = &sR[ii * 144 + lh * 3];
      const float xv0 = xr0[ii];
      const float xv1 = xr1[ii];
#pragma unroll
      for (int s = 0; s < 24; ++s) {
        const float4 q0 = rr[6 * s + 0];
        const float4 q1 = rr[6 * s + 1];
        const float4 q2 = rr[6 * s + 2];
        const v2f a0 = basis_cf_pair(xv0, q0, q1, q2);
        const v2f a1 = basis_cf_pair(xv1, q0, q1, q2);
        acc0 = __builtin_amdgcn_wmma_f32_16x16x4_f32(
            false, a0, false, bp[s], (short)0, acc0, false, false);
        acc1 = __builtin_amdgcn_wmma_f32_16x16x4_f32(
            false, a1, false, bp[s], (short)0, acc1, false, false);
      }
    }
  }

  // C/D layout: VGPR v -> row M = v + 8*lh, col N = m.
#pragma unroll
  for (int v = 0; v < 8; ++v) {
    out[(size_t)(nb + v + 8 * lh) * O + (ob * 16 + m)] = acc0[v];
    out[(size_t)(nb + 16 + v + 8 * lh) * O + (ob * 16 + m)] = acc1[v];
  }
}

extern "C" void kernel_launch(void* const* d_in, const int* in_sizes, int n_in,
                              void* d_out, int out_size, void* d_ws,
                              size_t ws_size, hipStream_t stream) {
  const float* x       = (const float*)d_in[0];  // (N, 64)
  const float* knots1  = (const float*)d_in[1];  // (128, 64, 10)
  const float* coeffs1 = (const float*)d_in[2];  // (128, 64, 10)
  const float* knots2  = (const float*)d_in[3];  // (64, 128, 10)
  const float* coeffs2 = (const float*)d_in[4];  // (64, 128, 10)
  float* out = (float*)d_out;                    // (N, 64)

  const int L0 = 64, L1 = 128, L2 = 64;
  const int N = in_sizes[0] / L0;                // 2048

  // Workspace layout (bytes):
  //   recA1: (128/16)*64*48*12*4 = 1179648
  //   recA2: (64/16)*128*48*12*4 = 1179648
  //   h    : N*128*4             = 1048576
  char* ws = (char*)d_ws;
  float* recA1 = (float*)(ws);
  float* recA2 = (float*)(ws + 1179648);
  float* h     = (float*)(ws + 2 * 1179648);

  {
    int tot1 = L1 * L0 * KJ;  // 49152
    int tot2 = L2 * L1 * KJ;  // 49152
    kan_prep<<<(tot1 + 255) / 256, 256, 0, stream>>>(knots1, coeffs1, recA1,
                                                     L1, L0);
    kan_prep<<<(tot2 + 255) / 256, 256, 0, stream>>>(knots2, coeffs2, recA2,
                                                     L2, L1);
  }

  // Layer 1: x (N x 64) -> h (N x 128)
  kan_layer<128, 64, 16><<<dim3(N / 32, 128 / 16), 32, 0, stream>>>(
      x, (const float4*)recA1, h);
  // Layer 2: h (N x 128) -> out (N x 64)
  kan_layer<64, 128, 16><<<dim3(N / 32, 64 / 16), 32, 0, stream>>>(
      h, (const float4*)recA2, out);
}